// Encoder_1245540516296
// MI455X (gfx1250) — compile-verified
//
#include <hip/hip_runtime.h>

typedef __attribute__((ext_vector_type(2))) float v2f;
typedef __attribute__((ext_vector_type(8))) float v8f;

#define N_NODES 50000
#define IN_FEATS 128
#define H_FEATS 64
#define N_EDGES 800000

// ---------------------------------------------------------------------------
// Vectorized zero (b128 stores). n must be a multiple of 4 (all our uses are).
// ---------------------------------------------------------------------------
__global__ void zero_f32_v4(float4* __restrict__ p, int n4) {
    int i = blockIdx.x * blockDim.x + threadIdx.x;
    const int stride = gridDim.x * blockDim.x;
    const float4 z = {0.0f, 0.0f, 0.0f, 0.0f};
    for (; i < n4; i += stride) p[i] = z;
}

// ---------------------------------------------------------------------------
// Degree accumulation: deg[dst[e]] += 1
// ---------------------------------------------------------------------------
__global__ void deg_accum(const int* __restrict__ dst, float* __restrict__ deg, int ne) {
    int i = blockIdx.x * blockDim.x + threadIdx.x;
    if (i < ne) atomicAdd(&deg[dst[i]], 1.0f);
}

// deg -> clip(deg,1)^-0.5 in place
__global__ void deg_to_dinv(float* __restrict__ deg, int n) {
    int i = blockIdx.x * blockDim.x + threadIdx.x;
    if (i < n) deg[i] = rsqrtf(fmaxf(deg[i], 1.0f));
}

// ---------------------------------------------------------------------------
// GEMM1: X = relu(F(50000x128) @ W1(64x128)^T + b1) via V_WMMA_F32_16X16X4_F32
// Block = 128 threads = 4 waves; block owns 16 rows, wave w owns N-tile
// [16w,16w+16). A tile staged once in LDS (row stride 132 floats: row base
// advances 4 banks/row -> the 16-lane column read is conflict-free), removing
// the 4x redundant global A loads. 50000 = 16*3125 exactly -> EXEC all-1s.
// ---------------------------------------------------------------------------
__global__ void gemm1_wmma(const float* __restrict__ F, const float* __restrict__ W1,
                           const float* __restrict__ b1, float* __restrict__ X) {
    __shared__ float shA[16][IN_FEATS + 4];          // 16 x 132 floats, 8.45 KB
    const int m0 = blockIdx.x * 16;

    // Cooperative A-tile load: 16x128 floats = 512 float4 chunks / 128 threads
#pragma unroll
    for (int c = threadIdx.x; c < 512; c += 128) {
        const int row = c >> 5;                      // 32 float4 per row
        const int col = (c & 31) << 2;
        const float4 v = *(const float4*)(F + (m0 + row) * IN_FEATS + col);
        *(float4*)(&shA[row][col]) = v;
    }
    __syncthreads();

    const int wave = threadIdx.x >> 5;
    const int lane = threadIdx.x & 31;
    const int n0 = wave * 16;
    const int half = lane >> 4;
    const int r = lane & 15;

    v8f acc = {};
    const float* brow = W1 + (n0 + r) * IN_FEATS + 2 * half;
#pragma unroll
    for (int k0 = 0; k0 < IN_FEATS; k0 += 4) {
        v2f a, b;
        a.x = shA[r][k0 + 2 * half];
        a.y = shA[r][k0 + 2 * half + 1];
        b.x = brow[k0 + 0];
        b.y = brow[k0 + 1];
        acc = __builtin_amdgcn_wmma_f32_16x16x4_f32(false, a, false, b,
                                                    (short)0, acc, false, false);
    }
    const float bias = b1[n0 + r];
#pragma unroll
    for (int v = 0; v < 8; ++v) {
        const int row = m0 + v + 8 * half;           // C/D layout: VGPR v, half
        const float val = acc[v] + bias;
        X[row * H_FEATS + n0 + r] = fmaxf(val, 0.0f);
    }
}

// ---------------------------------------------------------------------------
// Edge scatter: agg[dst[e]][:] += f[src[e]][:] * dinv[src[e]]
// One thread = 4 consecutive features of one edge: b128 gather load,
// 4 contiguous non-returning 32-bit float atomics (hardware floor).
// 16 consecutive threads cover one 256B edge row.
// ---------------------------------------------------------------------------
__global__ void edge_scatter(const float* __restrict__ f, const float* __restrict__ dinv,
                             const int* __restrict__ src, const int* __restrict__ dst,
                             float* __restrict__ agg) {
    const int total = N_EDGES * 16;                  // 12.8M quarter-rows
    int i = blockIdx.x * blockDim.x + threadIdx.x;
    const int stride = gridDim.x * blockDim.x;
    for (; i < total; i += stride) {
        const int e = i >> 4;
        const int q = (i & 15) << 2;                 // feature offset 0..60
        const int s = src[e];
        const int d = dst[e];
        const float4 v = *(const float4*)(f + s * H_FEATS + q);
        const float sc = dinv[s];
        float* ap = agg + d * H_FEATS + q;
        atomicAdd(ap + 0, v.x * sc);
        atomicAdd(ap + 1, v.y * sc);
        atomicAdd(ap + 2, v.z * sc);
        atomicAdd(ap + 3, v.w * sc);
    }
}

// f_out = f - agg * dinv[row], fully vectorized (b128 in/out)
__global__ void laplacian_combine(const float* __restrict__ f, const float* __restrict__ agg,
                                  const float* __restrict__ dinv, float* __restrict__ fout) {
    const int total = N_NODES * 16;                  // quarter-rows
    int i = blockIdx.x * blockDim.x + threadIdx.x;
    const int stride = gridDim.x * blockDim.x;
    for (; i < total; i += stride) {
        const int node = i >> 4;
        const int q = (i & 15) << 2;
        const float4 fv = *(const float4*)(f + node * H_FEATS + q);
        const float4 av = *(const float4*)(agg + node * H_FEATS + q);
        const float sc = dinv[node];
        float4 o;
        o.x = fv.x - av.x * sc;
        o.y = fv.y - av.y * sc;
        o.z = fv.z - av.z * sc;
        o.w = fv.w - av.w * sc;
        *(float4*)(fout + node * H_FEATS + q) = o;
    }
}

// ---------------------------------------------------------------------------
// Fold thetas into W2: W2eff_k[n][j] = sum_p THETA[p][k] * W2[n][p*64+j]
// THETA (from calc_thetas(2)): [[3,-3,0.75],[0,3,-1.5],[0,0,0.75]]
// ---------------------------------------------------------------------------
__global__ void make_w2eff(const float* __restrict__ W2, float* __restrict__ W2e) {
    const int idx = blockIdx.x * blockDim.x + threadIdx.x;
    if (idx >= 3 * H_FEATS * H_FEATS) return;
    const int k = idx / (H_FEATS * H_FEATS);
    const int n = (idx / H_FEATS) % H_FEATS;
    const int j = idx % H_FEATS;
    const float TH[3][3] = {{3.0f, -3.0f, 0.75f},
                            {0.0f,  3.0f, -1.5f},
                            {0.0f,  0.0f, 0.75f}};
    float v = 0.0f;
#pragma unroll
    for (int p = 0; p < 3; ++p)
        v += TH[p][k] * W2[n * (3 * H_FEATS) + p * H_FEATS + j];
    W2e[idx] = v;
}

// ---------------------------------------------------------------------------
// GEMM2: OUT = X@W2e0^T + F1@W2e1^T + F2@W2e2^T + b2 (single accumulator).
// All three 16x64 A tiles staged in LDS (row stride 68 floats -> 4 banks/row).
// ---------------------------------------------------------------------------
__global__ void gemm2_wmma(const float* __restrict__ X, const float* __restrict__ F1,
                           const float* __restrict__ F2, const float* __restrict__ W2e,
                           const float* __restrict__ b2, float* __restrict__ OUT) {
    __shared__ float shA[3][16][H_FEATS + 4];        // 3 x 16 x 68 floats, 13 KB
    const int m0 = blockIdx.x * 16;

    const float* bufs[3] = {X, F1, F2};
    // 3*16*64 floats = 768 float4 chunks / 128 threads
#pragma unroll
    for (int c = threadIdx.x; c < 768; c += 128) {
        const int kb = c >> 8;                       // 256 chunks per buffer
        const int row = (c >> 4) & 15;
        const int col = (c & 15) << 2;
        const float4 v = *(const float4*)(bufs[kb] + (m0 + row) * H_FEATS + col);
        *(float4*)(&shA[kb][row][col]) = v;
    }
    __syncthreads();

    const int wave = threadIdx.x >> 5;
    const int lane = threadIdx.x & 31;
    const int n0 = wave * 16;
    const int half = lane >> 4;
    const int r = lane & 15;

    v8f acc = {};
#pragma unroll
    for (int kb = 0; kb < 3; ++kb) {
        const float* brow = W2e + kb * H_FEATS * H_FEATS + (n0 + r) * H_FEATS + 2 * half;
#pragma unroll
        for (int k0 = 0; k0 < H_FEATS; k0 += 4) {
            v2f a, b;
            a.x = shA[kb][r][k0 + 2 * half];
            a.y = shA[kb][r][k0 + 2 * half + 1];
            b.x = brow[k0 + 0];
            b.y = brow[k0 + 1];
            acc = __builtin_amdgcn_wmma_f32_16x16x4_f32(false, a, false, b,
                                                        (short)0, acc, false, false);
        }
    }
    const float bias = b2[n0 + r];
#pragma unroll
    for (int v = 0; v < 8; ++v) {
        const int row = m0 + v + 8 * half;
        OUT[row * H_FEATS + n0 + r] = acc[v] + bias;
    }
}

// ---------------------------------------------------------------------------
// Launcher
// ---------------------------------------------------------------------------
extern "C" void kernel_launch(void* const* d_in, const int* in_sizes, int n_in,
                              void* d_out, int out_size, void* d_ws, size_t ws_size,
                              hipStream_t stream) {
    const float* features = (const float*)d_in[0];
    const int*   src      = (const int*)d_in[1];
    const int*   dst      = (const int*)d_in[2];
    const float* W1       = (const float*)d_in[3];
    const float* b1       = (const float*)d_in[4];
    const float* W2       = (const float*)d_in[5];
    const float* b2       = (const float*)d_in[6];
    float* out = (float*)d_out;

    // Workspace layout (floats)
    float* ws = (float*)d_ws;
    const size_t NH = (size_t)N_NODES * H_FEATS;        // 3.2M
    float* X    = ws;             // x = relu(...)
    float* F1v  = X    + NH;      // L(x)
    float* F2v  = F1v  + NH;      // L^2(x)
    float* AGG  = F2v  + NH;      // scatter accumulator
    float* DINV = AGG  + NH;      // degree -> D^-1/2 (50000)
    float* W2E  = DINV + N_NODES; // folded W2 (3*64*64)

    const int nh4 = (int)(NH / 4);                       // 800,000 float4
    const int rowBlocks = N_NODES / 16;                  // 3125 (exact)
    const int edgeQuads = N_EDGES * 16;                  // 12.8M
    const int edgeGrid  = edgeQuads / 256;               // 50,000

    // Degree / D^-1/2
    zero_f32_v4<<<(N_NODES / 4 + 255) / 256, 256, 0, stream>>>((float4*)DINV, N_NODES / 4);
    deg_accum<<<(N_EDGES + 255) / 256, 256, 0, stream>>>(dst, DINV, N_EDGES);
    deg_to_dinv<<<(N_NODES + 255) / 256, 256, 0, stream>>>(DINV, N_NODES);

    // Layer 1 GEMM (WMMA f32, LDS-staged A tile)
    gemm1_wmma<<<rowBlocks, 128, 0, stream>>>(features, W1, b1, X);

    // f1 = L(x)
    zero_f32_v4<<<(nh4 + 255) / 256, 256, 0, stream>>>((float4*)AGG, nh4);
    edge_scatter<<<edgeGrid, 256, 0, stream>>>(X, DINV, src, dst, AGG);
    laplacian_combine<<<(N_NODES * 16 + 255) / 256, 256, 0, stream>>>(X, AGG, DINV, F1v);

    // f2 = L(f1)
    zero_f32_v4<<<(nh4 + 255) / 256, 256, 0, stream>>>((float4*)AGG, nh4);
    edge_scatter<<<edgeGrid, 256, 0, stream>>>(F1v, DINV, src, dst, AGG);
    laplacian_combine<<<(N_NODES * 16 + 255) / 256, 256, 0, stream>>>(F1v, AGG, DINV, F2v);

    // Fold thetas into W2 and run final accumulated GEMM (WMMA f32)
    make_w2eff<<<(3 * H_FEATS * H_FEATS + 255) / 256, 256, 0, stream>>>(W2, W2E);
    gemm2_wmma<<<rowBlocks, 128, 0, stream>>>(X, F1v, F2v, W2E, b2, out);
}